// MemoryBank_31421980738217
// MI455X (gfx1250) — compile-verified
//
#include <hip/hip_runtime.h>

#define B_Q    4096
#define M_MEM  65536
#define D_DIM  512
#define TOPK   8
#define EPSF   1e-8f

#define BR 64            // queries per block
#define BC 128           // memory cols per tile
#define KC 32            // K-chunk staged in LDS
#define LDT 36           // LDS tile row stride (floats): 16B-aligned rows, conflict-free frag reads
#define LDS_S 129        // sims row stride (floats): conflict-free column scans
#define NCHUNK 8
#define COLS_PER_CHUNK (M_MEM / NCHUNK)   // 8192
#define NTILE (COLS_PER_CHUNK / BC)       // 64

typedef __attribute__((ext_vector_type(2))) float v2f;
typedef __attribute__((ext_vector_type(4))) float v4f;
typedef __attribute__((ext_vector_type(8))) float v8f;

// Sorted-descending top-k register list insert (fully unrolled bubble; no
// dynamic indexing so tv/ti stay in VGPRs). Strict '>' keeps earlier (lower
// index) entries on ties, matching lax.top_k.
__device__ __forceinline__ void topk_insert(float v, int idx,
                                            float (&tv)[TOPK], int (&ti)[TOPK]) {
  if (v > tv[TOPK - 1]) {
    float cv = v; int ci = idx;
#pragma unroll
    for (int p = 0; p < TOPK; ++p) {
      bool sw = cv > tv[p];
      float ov = tv[p]; int oi = ti[p];
      tv[p] = sw ? cv : ov;  ti[p] = sw ? ci : oi;
      cv    = sw ? ov : cv;  ci    = sw ? oi : ci;
    }
  }
}

// ---------------- Kernel 1: reciprocal row norms of the memory bank ----------
__global__ __launch_bounds__(256) void rnorm_kernel(const float* __restrict__ Mem,
                                                    float* __restrict__ rnorm) {
  const int lane = threadIdx.x & 31;
  const int row  = (blockIdx.x * 256 + threadIdx.x) >> 5;   // one wave per row
  const float* rowp = Mem + (size_t)row * D_DIM;
  float s = 0.f;
#pragma unroll
  for (int i = 0; i < 4; ++i) {
    v4f d = *(const v4f*)(rowp + lane * 4 + i * 128);
    s += d.x * d.x + d.y * d.y + d.z * d.z + d.w * d.w;
  }
#pragma unroll
  for (int off = 16; off >= 1; off >>= 1) s += __shfl_xor(s, off, 32);
  if (lane == 0) rnorm[row] = 1.0f / fmaxf(sqrtf(s), EPSF);
}

// ------------- Kernel 2: fused f32-WMMA similarity GEMM + running top-8 ------
__global__ __launch_bounds__(256) void gemm_topk_kernel(
    const float* __restrict__ Q, const float* __restrict__ Mem,
    const float* __restrict__ rnorm,
    float* __restrict__ cand_v, int* __restrict__ cand_i) {
  __shared__ float lQ[BR * LDT];        //  9216 B
  __shared__ float lM[BC * LDT];        // 18432 B
  __shared__ float lS[BR * LDS_S];      // 33024 B

  const int t    = threadIdx.x;
  const int lane = t & 31;
  const int wid  = t >> 5;              // 8 waves
  const int qbase = (wid >> 1) * 16;    // wave's query-tile (0,16,32,48)
  const int cbase = (wid & 1) * 64;     // wave's col-slab (0 or 64)
  const int l15  = lane & 15;
  const int lh   = lane >> 4;           // half-wave select (K split)

  const int qrow0 = blockIdx.y * BR;
  const int chunk = blockIdx.x;
  const int col_chunk0 = chunk * COLS_PER_CHUNK;

  // 4 scanner threads per query keep private running top-8 across all tiles
  const int q_local = t >> 2;           // 0..63
  const int sub     = t & 3;            // 0..3 -> 32-col span
  float tv[TOPK]; int ti[TOPK];
#pragma unroll
  for (int j = 0; j < TOPK; ++j) { tv[j] = -3.4e38f; ti[j] = 0; }

  for (int tile = 0; tile < NTILE; ++tile) {
    const int col0 = col_chunk0 + tile * BC;
    v8f acc0 = {}, acc1 = {}, acc2 = {}, acc3 = {};

    for (int kc = 0; kc < D_DIM / KC; ++kc) {
      const int kbase = kc * KC;
      __syncthreads();                       // prior LDS consumers done
#pragma unroll
      for (int i = 0; i < 2; ++i) {          // Q tile: 64x32 floats
        int idx = t + i * 256;
        int r = idx >> 3, c4 = (idx & 7) << 2;
        *(v4f*)(lQ + r * LDT + c4) =
            *(const v4f*)(Q + (size_t)(qrow0 + r) * D_DIM + kbase + c4);
      }
#pragma unroll
      for (int i = 0; i < 4; ++i) {          // Mem tile: 128x32 floats
        int idx = t + i * 256;
        int r = idx >> 3, c4 = (idx & 7) << 2;
        *(v4f*)(lM + r * LDT + c4) =
            *(const v4f*)(Mem + (size_t)(col0 + r) * D_DIM + kbase + c4);
      }
      __syncthreads();                       // tiles ready
#pragma unroll
      for (int ks = 0; ks < KC / 4; ++ks) {
        const int k0 = ks * 4 + 2 * lh;      // ISA A/B f32 frag layout
        v2f a  = *(const v2f*)(lQ + (qbase + l15) * LDT + k0);
        v2f b0 = *(const v2f*)(lM + (cbase +  0 + l15) * LDT + k0);
        v2f b1 = *(const v2f*)(lM + (cbase + 16 + l15) * LDT + k0);
        v2f b2 = *(const v2f*)(lM + (cbase + 32 + l15) * LDT + k0);
        v2f b3 = *(const v2f*)(lM + (cbase + 48 + l15) * LDT + k0);
        acc0 = __builtin_amdgcn_wmma_f32_16x16x4_f32(false, a, false, b0, (short)0, acc0, false, false);
        acc1 = __builtin_amdgcn_wmma_f32_16x16x4_f32(false, a, false, b1, (short)0, acc1, false, false);
        acc2 = __builtin_amdgcn_wmma_f32_16x16x4_f32(false, a, false, b2, (short)0, acc2, false, false);
        acc3 = __builtin_amdgcn_wmma_f32_16x16x4_f32(false, a, false, b3, (short)0, acc3, false, false);
      }
    }

    // Scale by per-memory-row reciprocal norm (column scale) and dump to LDS
    {
      float s0 = rnorm[col0 + cbase +  0 + l15];
      float s1 = rnorm[col0 + cbase + 16 + l15];
      float s2 = rnorm[col0 + cbase + 32 + l15];
      float s3 = rnorm[col0 + cbase + 48 + l15];
#pragma unroll
      for (int r = 0; r < 8; ++r) {
        int row = qbase + r + 8 * lh;        // C/D VGPR layout
        lS[row * LDS_S + cbase +  0 + l15] = acc0[r] * s0;
        lS[row * LDS_S + cbase + 16 + l15] = acc1[r] * s1;
        lS[row * LDS_S + cbase + 32 + l15] = acc2[r] * s2;
        lS[row * LDS_S + cbase + 48 + l15] = acc3[r] * s3;
      }
    }
    __syncthreads();

    // Running top-8 update: each thread scans its 32-col span of its query
    {
      const float* srow = lS + q_local * LDS_S + sub * 32;
      const int gbase = col0 + sub * 32;
#pragma unroll 4
      for (int c = 0; c < 32; ++c) topk_insert(srow[c], gbase + c, tv, ti);
    }
    __syncthreads();
  }

  // Merge the 4 partial lists per query; emit per-chunk top-8 candidates
  float* pv = lS;                 // 2048 floats
  int*   pi = (int*)lM;           // 2048 ints
#pragma unroll
  for (int j = 0; j < TOPK; ++j) { pv[t * TOPK + j] = tv[j]; pi[t * TOPK + j] = ti[j]; }
  __syncthreads();
  if (t < BR) {
    float fv[TOPK]; int fi[TOPK];
#pragma unroll
    for (int j = 0; j < TOPK; ++j) { fv[j] = -3.4e38f; fi[j] = 0; }
    for (int s = 0; s < 4; ++s)
#pragma unroll
      for (int j = 0; j < TOPK; ++j)
        topk_insert(pv[(t * 4 + s) * TOPK + j], pi[(t * 4 + s) * TOPK + j], fv, fi);
    const size_t base = (size_t)(qrow0 + t) * (NCHUNK * TOPK) + chunk * TOPK;
#pragma unroll
    for (int j = 0; j < TOPK; ++j) { cand_v[base + j] = fv[j]; cand_i[base + j] = fi[j]; }
  }
}

// ---------- Kernel 3: final merge of 64 candidates + gather/average ----------
__global__ __launch_bounds__(256) void merge_gather_kernel(
    const float* __restrict__ Mem, const float* __restrict__ cand_v,
    const int* __restrict__ cand_i, float* __restrict__ out) {
  __shared__ int sidx[TOPK];
  const int q = blockIdx.x;
  if (threadIdx.x == 0) {
    float fv[TOPK]; int fi[TOPK];
#pragma unroll
    for (int j = 0; j < TOPK; ++j) { fv[j] = -3.4e38f; fi[j] = 0; }
    const size_t base = (size_t)q * (NCHUNK * TOPK);
    for (int c = 0; c < NCHUNK * TOPK; ++c)
      topk_insert(cand_v[base + c], cand_i[base + c], fv, fi);
#pragma unroll
    for (int j = 0; j < TOPK; ++j) sidx[j] = fi[j];
  }
  __syncthreads();
#pragma unroll
  for (int i = 0; i < 2; ++i) {
    const int c = threadIdx.x + i * 256;
    float s = 0.f;
#pragma unroll
    for (int j = 0; j < TOPK; ++j) s += Mem[(size_t)sidx[j] * D_DIM + c];
    out[(size_t)q * D_DIM + c] = s * (1.0f / TOPK);
  }
}

extern "C" void kernel_launch(void* const* d_in, const int* in_sizes, int n_in,
                              void* d_out, int out_size, void* d_ws, size_t ws_size,
                              hipStream_t stream) {
  const float* Q   = (const float*)d_in[0];   // [4096, 512]
  const float* Mem = (const float*)d_in[1];   // [65536, 512]
  // d_in[2] is k (== 8), baked in as TOPK.

  float* rnorm  = (float*)d_ws;                                   // 65536 f
  float* cand_v = rnorm + M_MEM;                                  // 4096*64 f
  int*   cand_i = (int*)(cand_v + (size_t)B_Q * NCHUNK * TOPK);   // 4096*64 i
  float* out    = (float*)d_out;

  rnorm_kernel<<<M_MEM / 8, 256, 0, stream>>>(Mem, rnorm);
  dim3 g(NCHUNK, B_Q / BR);
  gemm_topk_kernel<<<g, 256, 0, stream>>>(Q, Mem, rnorm, cand_v, cand_i);
  merge_gather_kernel<<<B_Q, 256, 0, stream>>>(Mem, cand_v, cand_i, out);
}